// GIN_59055800320835
// MI455X (gfx1250) — compile-verified
//
#include <hip/hip_runtime.h>
#include <cstddef>

#define NNODES  100000
#define NEDGES  1250000
#define HIDC    64
#define OUTC    16
#define NGRAPHS 512

typedef float v2f __attribute__((ext_vector_type(2)));
typedef float v4f __attribute__((ext_vector_type(4)));
typedef float v8f __attribute__((ext_vector_type(8)));

// ---------------------------------------------------------------------------
// Y[nrows x 64] = relu( (X1 (+ X2)) @ W[64x64] + bias )
// One wave per 16-row tile; V_WMMA_F32_16X16X4_F32, 64 WMMAs per tile.
// A layout (32-bit A 16x4): lanes 0-15 -> K=k0,k0+1 ; lanes 16-31 -> K=k0+2,k0+3
// B layout mirrored; C/D: VGPR v -> row v (lanes 0-15) / v+8 (lanes 16-31).
// W is staged TRANSPOSED in LDS (Wt[n][k], padded stride) so every B fragment
// is one contiguous ds_load_b64 into an even-aligned VGPR pair (no repacking).
// ---------------------------------------------------------------------------
#define WT_STRIDE 66   // padded: avoids bank conflicts, keeps 8B alignment

__launch_bounds__(256, 2)
__global__ void gemm64_relu_wmma(const float* __restrict__ X1,
                                 const float* __restrict__ X2,
                                 const float* __restrict__ W,
                                 const float* __restrict__ bias,
                                 float* __restrict__ Y,
                                 int nrows)
{
    __shared__ float Wt[64 * WT_STRIDE];   // ~16.5 KB, transposed weights
    __shared__ float Xs_all[8 * 16 * 64];  // 32 KB (per-wave 16x64 tile)

    const int tid  = threadIdx.x;
    const int warp = tid >> 5;
    const int lane = tid & 31;
    const int half = lane >> 4;
    const int lm   = lane & 15;

    // Stage W transposed into LDS: Wt[n * WT_STRIDE + k] = W[k * 64 + n]
    #pragma unroll
    for (int i = 0; i < 16; ++i) {
        const int idx = i * 256 + tid;
        const int k = idx >> 6;
        const int n = idx & 63;
        Wt[n * WT_STRIDE + k] = W[idx];
    }
    __syncthreads();

    const int ntiles = nrows >> 4;                 // nrows % 16 == 0 here
    const int tile   = blockIdx.x * 8 + warp;
    if (tile >= ntiles) return;
    const size_t row0 = (size_t)tile << 4;

    float* Xs = Xs_all + warp * (16 * 64);
    const float* p1 = X1 + row0 * 64;
    if (X2 != nullptr) {                           // fused  z = h + agg
        const float* p2 = X2 + row0 * 64;
        #pragma unroll
        for (int i = 0; i < 8; ++i) {
            const int off = i * 128 + lane * 4;
            v4f a = *(const v4f*)(p1 + off);
            v4f c = *(const v4f*)(p2 + off);
            a.x += c.x; a.y += c.y; a.z += c.z; a.w += c.w;
            *(v4f*)(Xs + off) = a;
        }
    } else {
        #pragma unroll
        for (int i = 0; i < 8; ++i) {
            const int off = i * 128 + lane * 4;
            *(v4f*)(Xs + off) = *(const v4f*)(p1 + off);
        }
    }
    // Per-wave LDS write -> read handoff (DS ops are in-order per wave).
    asm volatile("s_wait_dscnt 0" ::: "memory");

    // Accumulators seeded with bias (bias depends only on the column)
    v8f acc0, acc1, acc2, acc3;
    {
        const float c0 = bias[lm], c1 = bias[16 + lm],
                    c2 = bias[32 + lm], c3 = bias[48 + lm];
        #pragma unroll
        for (int v = 0; v < 8; ++v) {
            acc0[v] = c0; acc1[v] = c1; acc2[v] = c2; acc3[v] = c3;
        }
    }

    const int ka = half << 1;                      // K sub-offset per lane half
    const float* xa  = Xs + lm * 64 + ka;
    const float* wt0 = Wt + (     lm) * WT_STRIDE + ka;
    const float* wt1 = Wt + (16 + lm) * WT_STRIDE + ka;
    const float* wt2 = Wt + (32 + lm) * WT_STRIDE + ka;
    const float* wt3 = Wt + (48 + lm) * WT_STRIDE + ka;

    #pragma unroll
    for (int k0 = 0; k0 < 64; k0 += 4) {
        const v2f a  = *(const v2f*)(xa  + k0);    // ds_load_b64 (pairs merge)
        const v2f b0 = *(const v2f*)(wt0 + k0);    // single ds_load_b64 each
        const v2f b1 = *(const v2f*)(wt1 + k0);
        const v2f b2 = *(const v2f*)(wt2 + k0);
        const v2f b3 = *(const v2f*)(wt3 + k0);
        acc0 = __builtin_amdgcn_wmma_f32_16x16x4_f32(false, a, false, b0, (short)0, acc0, false, false);
        acc1 = __builtin_amdgcn_wmma_f32_16x16x4_f32(false, a, false, b1, (short)0, acc1, false, false);
        acc2 = __builtin_amdgcn_wmma_f32_16x16x4_f32(false, a, false, b2, (short)0, acc2, false, false);
        acc3 = __builtin_amdgcn_wmma_f32_16x16x4_f32(false, a, false, b3, (short)0, acc3, false, false);
    }

    float* yp = Y + row0 * 64;
    #pragma unroll
    for (int v = 0; v < 8; ++v) {
        const int r = v + (half << 3);
        yp[r * 64 +      lm] = fmaxf(acc0[v], 0.0f);
        yp[r * 64 + 16 + lm] = fmaxf(acc1[v], 0.0f);
        yp[r * 64 + 32 + lm] = fmaxf(acc2[v], 0.0f);
        yp[r * 64 + 48 + lm] = fmaxf(acc3[v], 0.0f);
    }
}

// ---------------------------------------------------------------------------
// agg[dst] += h[src] : one thread per (edge, 4-channel group)
// ---------------------------------------------------------------------------
__global__ void scatter_add_kernel(const float* __restrict__ h,
                                   const int* __restrict__ src,
                                   const int* __restrict__ dst,
                                   float* __restrict__ agg)
{
    const int t = blockIdx.x * blockDim.x + threadIdx.x;
    const int e = t >> 4;
    if (e >= NEDGES) return;
    const int c4 = (t & 15) << 2;
    const int s = src[e];
    const int d = dst[e];
    const v4f v = *(const v4f*)(h + (size_t)s * HIDC + c4);
    float* p = agg + (size_t)d * HIDC + c4;
    atomicAdd(p + 0, v.x);
    atomicAdd(p + 1, v.y);
    atomicAdd(p + 2, v.z);
    atomicAdd(p + 3, v.w);
}

// ---------------------------------------------------------------------------
// Segment-sum pooling: sums[batch[n]] += h[n], cnt[batch[n]] += 1
// ---------------------------------------------------------------------------
__global__ void pool_kernel(const float* __restrict__ h,
                            const int* __restrict__ batch,
                            float* __restrict__ sums,
                            float* __restrict__ cnt)
{
    const int t = blockIdx.x * blockDim.x + threadIdx.x;
    if (t >= NNODES * HIDC) return;
    const int n = t >> 6;
    const int c = t & 63;
    const int g = batch[n];
    atomicAdd(&sums[g * HIDC + c], h[t]);
    if (c == 0) atomicAdd(&cnt[g], 1.0f);
}

// ---------------------------------------------------------------------------
// out[g, oc] = bl[oc] + sum_k (sums[g,k] / max(cnt[g],1)) * Wl[k,oc]
// ---------------------------------------------------------------------------
__global__ void final_kernel(const float* __restrict__ sums,
                             const float* __restrict__ cnt,
                             const float* __restrict__ Wl,
                             const float* __restrict__ bl,
                             float* __restrict__ out)
{
    const int t = blockIdx.x * blockDim.x + threadIdx.x;
    if (t >= NGRAPHS * OUTC) return;
    const int g  = t >> 4;
    const int oc = t & 15;
    const float inv = 1.0f / fmaxf(cnt[g], 1.0f);
    float acc = bl[oc];
    #pragma unroll 8
    for (int k = 0; k < HIDC; ++k)
        acc = fmaf(sums[g * HIDC + k] * inv, Wl[k * OUTC + oc], acc);
    out[t] = acc;
}

// ---------------------------------------------------------------------------
extern "C" void kernel_launch(void* const* d_in, const int* in_sizes, int n_in,
                              void* d_out, int out_size, void* d_ws, size_t ws_size,
                              hipStream_t stream)
{
    const float* x     = (const float*)d_in[0];
    const int*   eidx  = (const int*)d_in[1];   // [2, NEDGES]
    const int*   batch = (const int*)d_in[2];
    const float* W0 = (const float*)d_in[3];
    const float* b0 = (const float*)d_in[4];
    const float* Wa[3] = {(const float*)d_in[5],  (const float*)d_in[9],  (const float*)d_in[13]};
    const float* ba[3] = {(const float*)d_in[6],  (const float*)d_in[10], (const float*)d_in[14]};
    const float* Wb[3] = {(const float*)d_in[7],  (const float*)d_in[11], (const float*)d_in[15]};
    const float* bb[3] = {(const float*)d_in[8],  (const float*)d_in[12], (const float*)d_in[16]};
    const float* Wl = (const float*)d_in[17];
    const float* bl = (const float*)d_in[18];

    float* h    = (float*)d_ws;
    float* agg  = h    + (size_t)NNODES * HIDC;
    float* tmp  = agg  + (size_t)NNODES * HIDC;
    float* sums = tmp  + (size_t)NNODES * HIDC;
    float* cnt  = sums + NGRAPHS * HIDC;

    const int ntiles      = NNODES / 16;              // 6250, exact
    const int gemm_blocks = (ntiles + 7) / 8;         // 8 waves/block
    const int scat_blocks = (NEDGES * 16 + 255) / 256;

    // h = relu(x @ W0 + b0)
    gemm64_relu_wmma<<<gemm_blocks, 256, 0, stream>>>(x, nullptr, W0, b0, h, NNODES);

    const int* src = eidx;
    const int* dst = eidx + NEDGES;
    for (int l = 0; l < 3; ++l) {
        hipMemsetAsync(agg, 0, (size_t)NNODES * HIDC * sizeof(float), stream);
        scatter_add_kernel<<<scat_blocks, 256, 0, stream>>>(h, src, dst, agg);
        // tmp = relu((h + agg) @ Wa + ba) ; h = relu(tmp @ Wb + bb)
        gemm64_relu_wmma<<<gemm_blocks, 256, 0, stream>>>(h, agg, Wa[l], ba[l], tmp, NNODES);
        gemm64_relu_wmma<<<gemm_blocks, 256, 0, stream>>>(tmp, nullptr, Wb[l], bb[l], h, NNODES);
    }

    hipMemsetAsync(sums, 0, (NGRAPHS * HIDC + NGRAPHS) * sizeof(float), stream);
    pool_kernel<<<(NNODES * HIDC + 255) / 256, 256, 0, stream>>>(h, batch, sums, cnt);
    final_kernel<<<(NGRAPHS * OUTC + 255) / 256, 256, 0, stream>>>(sums, cnt, Wl, bl, (float*)d_out);
}